// HarmonicIntegral_63110249447948
// MI455X (gfx1250) — compile-verified
//
#include <hip/hip_runtime.h>
#include <hip/hip_bf16.h>

#define F_DIM 321
#define FP_DIM 324       // F padded to multiple of 4 (k-tail padding rows in LDS)
#define H_DIM 4200
#define T_DIM 1000
#define BC_DIM 16        // B*C = 8*2
#define KTOP 3
#define PAD_VAL 1e-8f
#define NEG_INF (-3.0e38f)

typedef float v2f __attribute__((ext_vector_type(2)));
typedef float v8f __attribute__((ext_vector_type(8)));
typedef unsigned int v4u __attribute__((ext_vector_type(4)));
typedef int v4i __attribute__((ext_vector_type(4)));
typedef int v8i __attribute__((ext_vector_type(8)));

__device__ __forceinline__ void top3_insert(float v, int idx,
                                            float& v0, int& i0,
                                            float& v1, int& i1,
                                            float& v2, int& i2) {
    if (v > v0)      { v2 = v1; i2 = i1; v1 = v0; i1 = i0; v0 = v; i0 = idx; }
    else if (v > v1) { v2 = v1; i2 = i1; v1 = v;  i1 = idx; }
    else if (v > v2) { v2 = v;  i2 = idx; }
}

#if defined(__has_builtin) && __has_builtin(__builtin_amdgcn_tensor_load_to_lds)
#define HAVE_TDM 1
// TDM: load a (F_DIM x 16)-column f32 tile of mag (row stride T_DIM) into LDS,
// row-major [f][16]. OOB t-columns (x >= td0) read as zero.
__device__ __forceinline__ void tdm_load_tile(const float* gbase, unsigned td0,
                                              const float* lds_dst) {
    const unsigned long long ga = (unsigned long long)(const void*)gbase;
    const unsigned lds_addr    = (unsigned)(unsigned long long)(const void*)lds_dst;
    v4u g0 = { 1u,                                   // count=1, user descriptor
               lds_addr,                             // lds_addr
               (unsigned)(ga & 0xFFFFFFFFu),         // global_addr[31:0]
               (unsigned)((ga >> 32) & 0x01FFFFFFu)  // global_addr[56:32]
                 | 0x80000000u };                    // type=2 ("image")
    v8i g1 = { (int)0x00020000u,                     // wg_mask=0, data_size=2 (4B)
               (int)((td0 & 0xFFFFu) << 16),         // tensor_dim0[15:0]
               (int)(((td0 >> 16) & 0xFFFFu)         // tensor_dim0[31:16]
                 | ((unsigned)F_DIM << 16)),         // tensor_dim1[15:0]
               (int)(16u << 16),                     // tensor_dim1[31:16]=0, tile_dim0=16
               (int)(unsigned)F_DIM,                 // tile_dim1=321, tile_dim2=0
               (int)(unsigned)T_DIM,                 // tensor_dim0_stride[31:0]=1000
               0, 0 };                               // stride0[47:32]=0, dim1_stride=0
    v4i g2 = { 0, 0, 0, 0 };
    v4i g3 = { 0, 0, 0, 0 };
    v8i g4 = { 0, 0, 0, 0, 0, 0, 0, 0 };
    __builtin_amdgcn_tensor_load_to_lds(g0, g1, g2, g3, g4, 0);
}
#else
#define HAVE_TDM 0
#endif

// Kernel 1: fused fp32 WMMA GEMM (integral_m @ mag) + running top-3 over H.
// grid: (ceil(T/32), B*C), block: 128 (4 waves).
// Two 16-column B tiles staged in LDS via the Tensor Data Mover; every A fetch
// feeds two v_wmma_f32_16x16x4_f32 (halves A traffic vs. a 16-wide block).
__global__ __launch_bounds__(128)
void harmonic_gemm_topk_kernel(const float* __restrict__ mag,      // (BC, F, T)
                               const float* __restrict__ integral, // (H, F)
                               int* __restrict__ pos)              // (BC, K, T)
{
    __shared__ float bt0[FP_DIM * 16];         // columns tbase .. tbase+15
    __shared__ float bt1[FP_DIM * 16];         // columns tbase+16 .. tbase+31
    __shared__ float sv[4][32][KTOP];
    __shared__ int   si[4][32][KTOP];

    const int tbase = blockIdx.x * 32;
    const int bc    = blockIdx.y;
    const int lane  = threadIdx.x & 31;
    const int wave  = threadIdx.x >> 5;
    const int n     = lane & 15;               // matrix column / A row owned by lane
    const int khalf = lane >> 4;               // 0: K 0..1 & rows 0..7, 1: K 2..3 & rows 8..15

    const float* __restrict__ magp = mag + (size_t)bc * F_DIM * T_DIM;

    // zero the k-padding rows f = 321..323 of both tiles
    if (threadIdx.x < (FP_DIM - F_DIM) * 16) {
        bt0[F_DIM * 16 + threadIdx.x] = 0.0f;
        bt1[F_DIM * 16 + threadIdx.x] = 0.0f;
    }

#if HAVE_TDM
    // second tile fully out of range only for the last t-block: zero it manually
    if (tbase + 16 >= T_DIM) {
        for (int idx = threadIdx.x; idx < F_DIM * 16; idx += 128)
            bt1[idx] = 0.0f;
    }
    if (wave == 0) {
        tdm_load_tile(magp + tbase, (unsigned)(T_DIM - tbase), bt0);
        if (tbase + 16 < T_DIM)
            tdm_load_tile(magp + tbase + 16, (unsigned)(T_DIM - tbase - 16), bt1);
        __builtin_amdgcn_s_wait_tensorcnt(0);
    }
#else
    for (int idx = threadIdx.x; idx < F_DIM * 16; idx += 128) {
        const int f  = idx >> 4;
        const int t0 = tbase + (idx & 15);
        const int t1 = t0 + 16;
        bt0[idx] = (t0 < T_DIM) ? magp[(size_t)f * T_DIM + t0] : 0.0f;
        bt1[idx] = (t1 < T_DIM) ? magp[(size_t)f * T_DIM + t1] : 0.0f;
    }
#endif
    __syncthreads();

    float a0v = NEG_INF, a1v = NEG_INF, a2v = NEG_INF;  // top-3, tile 0
    int   a0i = 0, a1i = 0, a2i = 0;
    float b0v = NEG_INF, b1v = NEG_INF, b2v = NEG_INF;  // top-3, tile 1
    int   b0i = 0, b1i = 0, b2i = 0;

    for (int hb = wave * 16; hb < H_DIM; hb += 64) {
        const int row = hb + n;
        const int rcl = (row < H_DIM) ? row : (H_DIM - 1);   // clamp; filtered below
        const float* __restrict__ arow = integral + (size_t)rcl * F_DIM;

        v8f acc0 = {}, acc1 = {};
        int k = 0;
        // main loop: fully unguarded (k+3 <= 319 < F); one A fetch, two WMMAs
        for (; k + 3 < F_DIM; k += 4) {
            const int ak = k + khalf * 2;
            v2f a, b0, b1;
            a.x  = arow[ak];
            a.y  = arow[ak + 1];
            b0.x = bt0[ak * 16 + n];
            b0.y = bt0[(ak + 1) * 16 + n];
            b1.x = bt1[ak * 16 + n];
            b1.y = bt1[(ak + 1) * 16 + n];
            acc0 = __builtin_amdgcn_wmma_f32_16x16x4_f32(
                       false, a, false, b0, (short)0, acc0, false, false);
            acc1 = __builtin_amdgcn_wmma_f32_16x16x4_f32(
                       false, a, false, b1, (short)0, acc1, false, false);
        }
        // k-tail (k == 320): B rows 321..323 are LDS zeros; only a.x needs a
        // select (a * 0 == 0 kills every other term); no EXEC manipulation.
        {
            const int ak = k + khalf * 2;
            const float alast = arow[F_DIM - 1];
            v2f a, b0, b1;
            a.x  = khalf ? 0.0f : alast;
            a.y  = 0.0f;
            b0.x = bt0[ak * 16 + n];
            b0.y = bt0[(ak + 1) * 16 + n];
            b1.x = bt1[ak * 16 + n];
            b1.y = bt1[(ak + 1) * 16 + n];
            acc0 = __builtin_amdgcn_wmma_f32_16x16x4_f32(
                       false, a, false, b0, (short)0, acc0, false, false);
            acc1 = __builtin_amdgcn_wmma_f32_16x16x4_f32(
                       false, a, false, b1, (short)0, acc1, false, false);
        }

        // acc{0,1}[r] = nominee[hb + r + 8*khalf, tbase + {0,16} + n]
#pragma unroll
        for (int r = 0; r < 8; ++r) {
            const int h = hb + r + 8 * khalf;
            const bool hv = (h < H_DIM);
            const float v0r = hv ? acc0[r] : NEG_INF;
            const float v1r = hv ? acc1[r] : NEG_INF;
            top3_insert(v0r, h, a0v, a0i, a1v, a1i, a2v, a2i);
            top3_insert(v1r, h, b0v, b0i, b1v, b1i, b2v, b2i);
        }
    }

    // merge the two half-wave lists for the same column (lane ^ 16)
    {
        float pv; int pi;
        pv = __shfl_xor(a0v, 16, 32); pi = __shfl_xor(a0i, 16, 32);
        float q1 = __shfl_xor(a1v, 16, 32); int qi1 = __shfl_xor(a1i, 16, 32);
        float q2 = __shfl_xor(a2v, 16, 32); int qi2 = __shfl_xor(a2i, 16, 32);
        top3_insert(pv, pi, a0v, a0i, a1v, a1i, a2v, a2i);
        top3_insert(q1, qi1, a0v, a0i, a1v, a1i, a2v, a2i);
        top3_insert(q2, qi2, a0v, a0i, a1v, a1i, a2v, a2i);

        pv = __shfl_xor(b0v, 16, 32); pi = __shfl_xor(b0i, 16, 32);
        q1 = __shfl_xor(b1v, 16, 32); qi1 = __shfl_xor(b1i, 16, 32);
        q2 = __shfl_xor(b2v, 16, 32); qi2 = __shfl_xor(b2i, 16, 32);
        top3_insert(pv, pi, b0v, b0i, b1v, b1i, b2v, b2i);
        top3_insert(q1, qi1, b0v, b0i, b1v, b1i, b2v, b2i);
        top3_insert(q2, qi2, b0v, b0i, b1v, b1i, b2v, b2i);
    }

    if (lane < 16) {
        sv[wave][n][0] = a0v;      si[wave][n][0] = a0i;
        sv[wave][n][1] = a1v;      si[wave][n][1] = a1i;
        sv[wave][n][2] = a2v;      si[wave][n][2] = a2i;
        sv[wave][16 + n][0] = b0v; si[wave][16 + n][0] = b0i;
        sv[wave][16 + n][1] = b1v; si[wave][16 + n][1] = b1i;
        sv[wave][16 + n][2] = b2v; si[wave][16 + n][2] = b2i;
    }
    __syncthreads();

    if (threadIdx.x < 32) {
        const int col = threadIdx.x;
        float m0 = NEG_INF, m1 = NEG_INF, m2 = NEG_INF;
        int   j0 = 0, j1 = 0, j2 = 0;
#pragma unroll
        for (int w = 0; w < 4; ++w)
#pragma unroll
            for (int j = 0; j < KTOP; ++j)
                top3_insert(sv[w][col][j], si[w][col][j], m0, j0, m1, j1, m2, j2);

        const int tt = tbase + col;
        if (tt < T_DIM) {
            int* __restrict__ pp = pos + (size_t)bc * KTOP * T_DIM + tt;
            pp[0]         = j0;
            pp[T_DIM]     = j1;
            pp[2 * T_DIM] = j2;
        }
    }
}

// Kernel 2: causal avg-pool of indices + gather harmonic_loc + sum + threshold.
// grid: (ceil(T/64), F, BC), block: 64 (coalesced along t for the output).
__global__ __launch_bounds__(64)
void harmonic_pool_gather_kernel(const int* __restrict__ pos,    // (BC, K, T)
                                 const float* __restrict__ hloc, // (H, F)
                                 float* __restrict__ out)        // (BC, F, T)
{
    const int t  = blockIdx.x * 64 + threadIdx.x;
    const int f  = blockIdx.y;
    const int bc = blockIdx.z;
    if (t >= T_DIM) return;

    float s = 0.0f;
#pragma unroll
    for (int k = 0; k < KTOP; ++k) {
        const int* __restrict__ pk = pos + ((size_t)bc * KTOP + k) * T_DIM;
        const float p0 = (float)pk[t];
        const float p1 = (t >= 1) ? (float)pk[t - 1] : PAD_VAL;
        const float p2 = (t >= 2) ? (float)pk[t - 2] : PAD_VAL;
        int choose = (int)((p0 + p1 + p2) / 3.0f);   // truncation, like astype(int32)
        choose = choose < 0 ? 0 : (choose >= H_DIM ? H_DIM - 1 : choose);
        s += hloc[(size_t)choose * F_DIM + f];
    }
    out[((size_t)bc * F_DIM + f) * T_DIM + t] = (s > 0.0f) ? 1.0f : 0.0f;
}

extern "C" void kernel_launch(void* const* d_in, const int* in_sizes, int n_in,
                              void* d_out, int out_size, void* d_ws, size_t ws_size,
                              hipStream_t stream) {
    const float* mag      = (const float*)d_in[0];  // (8,2,321,1000)
    const float* integral = (const float*)d_in[1];  // (4200,321)
    const float* hloc     = (const float*)d_in[2];  // (4200,321)
    float* out            = (float*)d_out;          // (8,2,321,1000)
    int*   pos            = (int*)d_ws;             // (16,3,1000) = 192 KB scratch

    dim3 g1((T_DIM + 31) / 32, BC_DIM);
    harmonic_gemm_topk_kernel<<<g1, 128, 0, stream>>>(mag, integral, pos);

    dim3 g2((T_DIM + 63) / 64, F_DIM, BC_DIM);
    harmonic_pool_gather_kernel<<<g2, 64, 0, stream>>>(pos, hloc, out);
}